// CCDet_45518063403068
// MI455X (gfx1250) — compile-verified
//
#include <hip/hip_runtime.h>
#include <hip/hip_bf16.h>
#include <stdint.h>

#define N_ANCH     102400
#define FMP        320
#define NUM_CLS    80
#define K_TOP      1000
#define K_PAD      1008          // 63 tiles of 16
#define NMS_THR    0.6f
#define SCL_CLAMP  6.907755278982137f   // log(1000)
#define BOX_SCL    0.003125f     // STRIDE / IMG_SIZE = 4/1280

typedef float v2f __attribute__((ext_vector_type(2)));
typedef float v8f __attribute__((ext_vector_type(8)));

// ---------------------------------------------------------------------------
// Pass 1: fused score + argmax per anchor (memory-bound: 32.8 MB stream).
// argmax_c fused == argmax_c hmp (sig monotone, sig(iou) per-anchor const>0),
// so only one exp per anchor is needed instead of 80.
// ---------------------------------------------------------------------------
__global__ void score_kernel(const float* __restrict__ hmp,
                             const float* __restrict__ iouP,
                             float* __restrict__ scores,
                             int*   __restrict__ labels) {
  int a = blockIdx.x * blockDim.x + threadIdx.x;
  if (a >= N_ANCH) return;
  const float4* row = (const float4*)(hmp + (size_t)a * NUM_CLS);
  float best = -3.4e38f; int bl = 0;
#pragma unroll
  for (int g = 0; g < NUM_CLS / 4; ++g) {
    if (g + 4 < NUM_CLS / 4) __builtin_prefetch(row + g + 4, 0, 0); // global_prefetch_b8
    float4 q = row[g];
    if (q.x > best) { best = q.x; bl = 4 * g + 0; }
    if (q.y > best) { best = q.y; bl = 4 * g + 1; }
    if (q.z > best) { best = q.z; bl = 4 * g + 2; }
    if (q.w > best) { best = q.w; bl = 4 * g + 3; }
  }
  float sh = 1.0f / (1.0f + expf(-best));
  float si = 1.0f / (1.0f + expf(-iouP[a]));
  scores[a] = sqrtf(sh * si);
  labels[a] = bl;
}

// ---------------------------------------------------------------------------
// Pass 2: exact top-1000 via 8x8-bit radix select on key=(scoreBits<<32)|~idx,
// then single-block bitonic sort (descending) of the 1000 winners.
// Key order reproduces lax.top_k: value desc, index asc on ties.
// ---------------------------------------------------------------------------
__global__ void topk_kernel(const float* __restrict__ scores,
                            const int*   __restrict__ labels,
                            int*   __restrict__ topIdx,
                            float* __restrict__ out) {
  __shared__ unsigned hist[256];
  __shared__ unsigned long long shPrefix;
  __shared__ int shK, shCnt;
  __shared__ unsigned long long ck[1024];
  __shared__ unsigned           ci[1024];
  int tid = threadIdx.x;
  if (tid == 0) { shPrefix = 0ull; shK = K_TOP; }
  __syncthreads();

  for (int r = 0; r < 8; ++r) {
    hist[tid] = 0u;
    __syncthreads();
    int shift = 56 - 8 * r;
    unsigned long long pfx = shPrefix;
    for (int a = tid; a < N_ANCH; a += 256) {
      unsigned long long key =
        ((unsigned long long)__float_as_uint(scores[a]) << 32) | (unsigned)(~a);
      bool match = (r == 0) || ((key >> (shift + 8)) == pfx);
      if (match) atomicAdd(&hist[(unsigned)((key >> shift) & 0xFFull)], 1u);
    }
    __syncthreads();
    if (tid == 0) {
      int K = shK; unsigned run = 0; int sel = 0;
      for (int d = 255; d >= 0; --d) {
        unsigned c = hist[d];
        if (run + c >= (unsigned)K) { sel = d; break; }
        run += c;
      }
      shK = K - (int)run;
      shPrefix = (pfx << 8) | (unsigned)sel;
    }
    __syncthreads();
  }
  unsigned long long kth = shPrefix;   // exact key of the 1000th largest
  if (tid == 0) shCnt = 0;
  __syncthreads();

  for (int a = tid; a < N_ANCH; a += 256) {
    unsigned long long key =
      ((unsigned long long)__float_as_uint(scores[a]) << 32) | (unsigned)(~a);
    if (key >= kth) {                  // keys distinct -> exactly 1000 hits
      int pos = atomicAdd(&shCnt, 1);
      if (pos < 1024) { ck[pos] = key; ci[pos] = (unsigned)a; }
    }
  }
  __syncthreads();
  for (int i = tid; i < 1024; i += 256)
    if (i >= shCnt) { ck[i] = 0ull; ci[i] = 0u; }
  __syncthreads();

  for (int ksz = 2; ksz <= 1024; ksz <<= 1) {
    for (int j = ksz >> 1; j > 0; j >>= 1) {
      for (int i = tid; i < 1024; i += 256) {
        int p = i ^ j;
        if (p > i) {
          bool dirA = ((i & ksz) == 0);
          unsigned long long a0 = ck[i], b0 = ck[p];
          if (dirA ? (a0 < b0) : (a0 > b0)) {   // descending overall
            ck[i] = b0; ck[p] = a0;
            unsigned t = ci[i]; ci[i] = ci[p]; ci[p] = t;
          }
        }
      }
      __syncthreads();
    }
  }
  for (int i = tid; i < K_TOP; i += 256) {
    unsigned long long key = ck[i];
    int a = (int)ci[i];
    out[i]          = __uint_as_float((unsigned)(key >> 32));  // scores
    out[K_TOP + i]  = (float)labels[a];                        // labels
    topIdx[i] = a;
  }
}

// ---------------------------------------------------------------------------
// Pass 3: decode + clip the 1000 boxes; pad 1000..1007 with inert entries
// (zero box / label -1 => can never suppress a real box).
// ---------------------------------------------------------------------------
__global__ void decode_kernel(const float* __restrict__ reg,
                              const int*   __restrict__ topIdx,
                              const int*   __restrict__ labels,
                              float4* __restrict__ boxes,
                              int*    __restrict__ boxLab,
                              float*  __restrict__ areas,
                              float*  __restrict__ out) {
  int i = blockIdx.x * blockDim.x + threadIdx.x;
  if (i >= K_PAD) return;
  if (i < K_TOP) {
    int a = topIdx[i];
    float ax = (float)(a % FMP), ay = (float)(a / FMP);
    float4 r4 = ((const float4*)reg)[a];
    float e0 = expf(fminf(r4.x, SCL_CLAMP));
    float e1 = expf(fminf(r4.y, SCL_CLAMP));
    float e2 = expf(fminf(r4.z, SCL_CLAMP));
    float e3 = expf(fminf(r4.w, SCL_CLAMP));
    float x1 = fminf(fmaxf((ax - e0) * BOX_SCL, 0.f), 1.f);
    float y1 = fminf(fmaxf((ay - e1) * BOX_SCL, 0.f), 1.f);
    float x2 = fminf(fmaxf((ax + e2) * BOX_SCL, 0.f), 1.f);
    float y2 = fminf(fmaxf((ay + e3) * BOX_SCL, 0.f), 1.f);
    boxes[i]  = make_float4(x1, y1, x2, y2);
    boxLab[i] = labels[a];
    areas[i]  = (x2 - x1) * (y2 - y1);
    out[2 * K_TOP + 4 * i + 0] = x1;
    out[2 * K_TOP + 4 * i + 1] = y1;
    out[2 * K_TOP + 4 * i + 2] = x2;
    out[2 * K_TOP + 4 * i + 3] = y2;
  } else {
    boxes[i]  = make_float4(0.f, 0.f, 0.f, 0.f);
    boxLab[i] = -1;
    areas[i]  = 0.f;
  }
}

// ---------------------------------------------------------------------------
// Pass 4: 16x16 IoU tiles. Union base term area_i + area_j is a rank-2 outer
// product computed on the matrix pipe: D = [area_i | 1] x [[1..1],[area_j]]
// via V_WMMA_F32_16X16X4_F32 (wave32 C/D layout: lane<16 -> M=v, else M=v+8).
// The IoU>thr test uses the multiply form (union>0 by construction), avoiding
// the IEEE v_div_scale/v_div_fmas/v_div_fixup chain per element.
// Suppression bits harvested with wave32 ballot into bitmask rows.
// ---------------------------------------------------------------------------
__global__ void iou_kernel(const float4* __restrict__ boxes,
                           const int*    __restrict__ boxLab,
                           const float*  __restrict__ areas,
                           unsigned*     __restrict__ suppress) {
  int ti = blockIdx.x, tj = blockIdx.y;
  int lane = threadIdx.x;
  __shared__ float4 rB[16], cB[16];
  __shared__ float  rA[16], cA[16];
  __shared__ int    rL[16], cL[16];
  if (lane < 16) {
    int r = ti * 16 + lane;
    rB[lane] = boxes[r]; rA[lane] = areas[r]; rL[lane] = boxLab[r];
  } else {
    int c = tj * 16 + (lane - 16);
    cB[lane - 16] = boxes[c]; cA[lane - 16] = areas[c]; cL[lane - 16] = boxLab[c];
  }
  __syncthreads();

  // A 16x4: lanes0-15 hold K=0(v0),K=1(v1); lanes16-31 hold K=2,K=3 (zeros).
  // B 4x16: v0 = row K=0 (ones) / K=2 (zeros); v1 = row K=1 (area_j) / K=3.
  v2f Af, Bf;
  Af.x = (lane < 16) ? rA[lane] : 0.0f;
  Af.y = (lane < 16) ? 1.0f     : 0.0f;
  Bf.x = (lane < 16) ? 1.0f     : 0.0f;
  Bf.y = (lane < 16) ? cA[lane] : 0.0f;
  v8f C = {};
  v8f D = __builtin_amdgcn_wmma_f32_16x16x4_f32(
      false, Af, false, Bf, (short)0, C, false, false);  // D[m][n]=area_i+area_j

  int rowBase = ti * 16;
  unsigned word  = (unsigned)(tj >> 1);
  unsigned shiftb = (unsigned)(tj & 1) * 16u;
  int n = lane & 15;
#pragma unroll
  for (int v = 0; v < 8; ++v) {
    int m = v + ((lane >= 16) ? 8 : 0);
    float4 bi = rB[m], bj = cB[n];
    float xx1 = fmaxf(bi.x, bj.x), yy1 = fmaxf(bi.y, bj.y);
    float xx2 = fminf(bi.z, bj.z), yy2 = fminf(bi.w, bj.w);
    float w = fmaxf(1e-10f, xx2 - xx1);
    float h = fmaxf(1e-10f, yy2 - yy1);
    float inter = w * h;
    float uni   = D[v] - inter + 1e-10f;   // always > 0
    bool sup = (inter > NMS_THR * uni) && (rL[m] == cL[n]); // iou > thr
    unsigned bal = (unsigned)(__ballot(sup) & 0xFFFFFFFFull);
    if (lane == 0)
      atomicOr(&suppress[(rowBase + v) * 32 + word], (bal & 0xFFFFu) << shiftb);
    if (lane == 16)
      atomicOr(&suppress[(rowBase + v + 8) * 32 + word], ((bal >> 16) & 0xFFFFu) << shiftb);
  }
}

// ---------------------------------------------------------------------------
// Pass 5: greedy NMS, single wave32 scan. Lane w owns keep bits [32w,32w+31].
// Suppress rows are independent of the keep state, so a depth-4 register
// pipeline keeps rows in flight across the serialized scan; keep[i] is
// broadcast with a wave32 shuffle.
// ---------------------------------------------------------------------------
__global__ void nms_kernel(const unsigned* __restrict__ sup,
                           float* __restrict__ out) {
  __shared__ unsigned keepW[32];
  int tid = threadIdx.x;
  if (tid < 32) {
    unsigned keep = 0xFFFFFFFFu;
    unsigned r0 = sup[0 * 32 + tid];
    unsigned r1 = sup[1 * 32 + tid];
    unsigned r2 = sup[2 * 32 + tid];
    unsigned r3 = sup[3 * 32 + tid];
    for (int i = 0; i < K_TOP; ++i) {
      unsigned rNew = (i + 4 < K_TOP) ? sup[(i + 4) * 32 + tid] : 0u;
      int wi = i >> 5, bi = i & 31;
      unsigned kw = (unsigned)__shfl((int)keep, wi, 32);
      if ((kw >> bi) & 1u) {
        unsigned gt = (tid > wi) ? 0xFFFFFFFFu
                    : ((tid == wi) ? (0xFFFFFFFEu << bi) : 0u); // bi=31 -> 0
        keep &= ~(r0 & gt);
      }
      r0 = r1; r1 = r2; r2 = r3; r3 = rNew;
    }
    keepW[tid] = keep;
  }
  __syncthreads();
  for (int j = tid; j < K_TOP; j += blockDim.x)
    out[6 * K_TOP + j] = (float)((keepW[j >> 5] >> (j & 31)) & 1u);
}

// ---------------------------------------------------------------------------
extern "C" void kernel_launch(void* const* d_in, const int* in_sizes, int n_in,
                              void* d_out, int out_size, void* d_ws, size_t ws_size,
                              hipStream_t stream) {
  const float* hmp  = (const float*)d_in[0];   // (102400, 80)
  const float* reg  = (const float*)d_in[1];   // (102400, 4)
  const float* iouP = (const float*)d_in[2];   // (102400, 1)
  float* out = (float*)d_out;                  // 7000 floats

  char* ws = (char*)d_ws;
  float*    scores   = (float*)   (ws + 0);        // 102400 f32
  int*      labels   = (int*)     (ws + 409600);   // 102400 i32
  int*      topIdx   = (int*)     (ws + 819200);   // 1024 i32
  float4*   boxes    = (float4*)  (ws + 823296);   // 1008 float4
  int*      boxLab   = (int*)     (ws + 839424);   // 1008 i32 (padded)
  float*    areas    = (float*)   (ws + 843520);   // 1008 f32 (padded)
  unsigned* suppress = (unsigned*)(ws + 847616);   // 1008*32 u32 bitmask

  hipMemsetAsync(suppress, 0, (size_t)K_PAD * 32 * sizeof(unsigned), stream);
  score_kernel <<<(N_ANCH + 255) / 256, 256, 0, stream>>>(hmp, iouP, scores, labels);
  topk_kernel  <<<1, 256, 0, stream>>>(scores, labels, topIdx, out);
  decode_kernel<<<(K_PAD + 255) / 256, 256, 0, stream>>>(reg, topIdx, labels,
                                                         boxes, boxLab, areas, out);
  iou_kernel   <<<dim3(63, 63), 32, 0, stream>>>(boxes, boxLab, areas, suppress);
  nms_kernel   <<<1, 1024, 0, stream>>>(suppress, out);
}